// SoftDotAttention_25786983645958
// MI455X (gfx1250) — compile-verified
//
#include <hip/hip_runtime.h>
#include <math.h>
#include <stdint.h>

typedef __attribute__((ext_vector_type(2))) float v2f;
typedef __attribute__((ext_vector_type(8))) float v8f;

#define B_   64
#define L_   1024
#define D_   2048
#define SEG  4
#define RSEG (L_ / SEG)        // 256 rows per segment
#define CJ   8                 // rows staged per chunk (8 x 8KB = 64KB)
#define NEG_BIG 1e12f

// ---- CDNA5 async global->LDS path (guarded; falls back to sync copy) ------
#if defined(__has_builtin)
#if __has_builtin(__builtin_amdgcn_global_load_async_to_lds_b128)
#define USE_ASYNC 1
#endif
#endif
#ifndef USE_ASYNC
#define USE_ASYNC 0
#endif

#if USE_ASYNC
// Builtin signature (from clang diagnostic): (int4 AS1*, int4 AS3*, imm, imm)
typedef __attribute__((__vector_size__(4 * sizeof(int)))) int i32x4;
typedef __attribute__((address_space(1))) i32x4 as1_i32x4;
typedef __attribute__((address_space(3))) i32x4 as3_i32x4;
#if __has_builtin(__builtin_amdgcn_s_wait_asynccnt)
#define WAIT_ASYNC(n) __builtin_amdgcn_s_wait_asynccnt(n)
#else
#define WAIT_ASYNC(n) asm volatile("s_wait_asynccnt %0" ::"i"(n) : "memory")
#endif
#else
#define WAIT_ASYNC(n) ((void)0)
#endif

// ---------------------------------------------------------------------------
// K1: target[64,2048] = input[64,2048] @ W_in[2048,2048]^T
// One wave per 16x16 tile, V_WMMA_F32_16X16X4_F32 over K=2048.
// ---------------------------------------------------------------------------
__global__ __launch_bounds__(32)
void gemm_in_kernel(const float* __restrict__ input,
                    const float* __restrict__ Win,
                    float* __restrict__ tgt) {
  const int lane = threadIdx.x;
  const int mn   = lane & 15;
  const int half = lane >> 4;
  const int n0   = blockIdx.x * 16;
  const int m0   = blockIdx.y * 16;

  const float* arow = input + (size_t)(m0 + mn) * D_;
  const float* brow = Win   + (size_t)(n0 + mn) * D_;

  v8f c = {};
  #pragma unroll 4
  for (int k0 = 0; k0 < D_; k0 += 4) {
    const int k = k0 + 2 * half;
    v2f a = *(const v2f*)(arow + k);
    v2f b = *(const v2f*)(brow + k);
    c = __builtin_amdgcn_wmma_f32_16x16x4_f32(false, a, false, b,
                                              (short)0, c, false, false);
  }
  #pragma unroll
  for (int r = 0; r < 8; ++r)
    tgt[(size_t)(m0 + r + 8 * half) * D_ + n0 + mn] = c[r];
}

// ---------------------------------------------------------------------------
// K2: fused logits + mask + online softmax + weighted accumulation.
// Grid (SEG, B), 256 threads (8 waves). Context is consumed in 64KB chunks
// staged into a double-buffered LDS tile (async global->LDS when available),
// so HBM sees exactly one pass over the 512MB context tensor. Both the
// per-row dot products and the weighted accumulation read the chunk from LDS
// with b128 accesses.
// ---------------------------------------------------------------------------
__global__ __launch_bounds__(256)
void attn_partial_kernel(const float* __restrict__ ctx,
                         const uint8_t* __restrict__ mask,
                         const float* __restrict__ tgt,
                         float* __restrict__ attn_raw,
                         float* __restrict__ mseg,
                         float* __restrict__ sseg,
                         float* __restrict__ accseg) {
  __shared__ float4 buf[2][CJ * (D_ / 4)];   // 2 x 64KB chunk buffers
  __shared__ float4 tgt4[D_ / 4];            // 8KB staged target row
  __shared__ float  logits[CJ];

  const int tid  = threadIdx.x;
  const int b    = blockIdx.y;
  const int seg  = blockIdx.x;
  const int jb   = seg * RSEG;
  const int wave = tid >> 5;
  const int lane = tid & 31;

  const float*  cb  = ctx + (size_t)b * L_ * D_;
  const float4* cb4 = (const float4*)cb;

  // stage target[b,:] into LDS
  {
    const float4* tg = (const float4*)(tgt + (size_t)b * D_);
    #pragma unroll
    for (int i = 0; i < (D_ / 4) / 256; ++i) tgt4[i * 256 + tid] = tg[i * 256 + tid];
  }

  // chunk staging: CJ*D_/4 = 4096 float4 over 256 threads -> 16 ops/thread
  auto stage = [&](int bi, int jc) {
    const float4* src = cb4 + (size_t)(jb + jc) * (D_ / 4);
    float4* dst = &buf[bi][0];
    #pragma unroll
    for (int q = 0; q < 16; ++q) {
      const int idx = q * 256 + tid;
#if USE_ASYNC
      __builtin_amdgcn_global_load_async_to_lds_b128(
          (as1_i32x4*)(src + idx), (as3_i32x4*)(dst + idx), 0, 0);
#else
      dst[idx] = src[idx];
#endif
    }
  };

  stage(0, 0);

  float m = -INFINITY;
  float s = 0.f;
  float4 a0 = {0.f, 0.f, 0.f, 0.f};
  float4 a1 = {0.f, 0.f, 0.f, 0.f};

  int cur = 0;
  for (int jc = 0; jc < RSEG; jc += CJ) {
    if (jc + CJ < RSEG) { stage(cur ^ 1, jc + CJ); WAIT_ASYNC(16); }
    else                { WAIT_ASYNC(0); }
    __syncthreads();   // chunk `cur` resident in LDS for all waves

    // --- one dot product per wave (row = wave) ---
    {
      const int j = jb + jc + wave;
      const float4* row = &buf[cur][wave * (D_ / 4)];
      float p = 0.f;
      #pragma unroll 4
      for (int i = 0; i < (D_ / 4) / 32; ++i) {      // 16 iters of b128
        const float4 cv = row[i * 32 + lane];
        const float4 tv = tgt4[i * 32 + lane];
        p += cv.x * tv.x + cv.y * tv.y + cv.z * tv.z + cv.w * tv.w;
      }
      #pragma unroll
      for (int off = 16; off > 0; off >>= 1) p += __shfl_down(p, off, 32);
      if (lane == 0) {
        const float lg = mask[(size_t)b * L_ + j] ? -NEG_BIG : p;
        logits[wave] = lg;
        attn_raw[(size_t)b * L_ + j] = lg;           // normalized later
      }
    }
    __syncthreads();

    // --- online softmax update ---
    float cm = logits[0];
    #pragma unroll
    for (int w = 1; w < CJ; ++w) cm = fmaxf(cm, logits[w]);
    const float mnew  = fmaxf(m, cm);
    const float scale = __expf(m - mnew);            // first iter: exp(-inf)=0
    s *= scale;
    a0.x *= scale; a0.y *= scale; a0.z *= scale; a0.w *= scale;
    a1.x *= scale; a1.y *= scale; a1.z *= scale; a1.w *= scale;

    #pragma unroll
    for (int w = 0; w < CJ; ++w) {
      const float pw = __expf(logits[w] - mnew);
      const float4 c0 = buf[cur][w * (D_ / 4) + tid];        // d = 4*tid
      const float4 c1 = buf[cur][w * (D_ / 4) + 256 + tid];  // d = 1024+4*tid
      a0.x += pw * c0.x; a0.y += pw * c0.y; a0.z += pw * c0.z; a0.w += pw * c0.w;
      a1.x += pw * c1.x; a1.y += pw * c1.y; a1.z += pw * c1.z; a1.w += pw * c1.w;
      s += pw;
    }
    m = mnew;
    __syncthreads();   // everyone done with buf[cur] & logits before reuse
    cur ^= 1;
  }

  if (tid == 0) { mseg[b * SEG + seg] = m; sseg[b * SEG + seg] = s; }
  float4* as4 = (float4*)(accseg + (size_t)(b * SEG + seg) * D_);
  as4[tid]       = a0;
  as4[256 + tid] = a1;
}

// ---------------------------------------------------------------------------
// K2b: merge SEG partial softmax states per batch; emit weighted_context and
// normalize attn logits in place.
// ---------------------------------------------------------------------------
__global__ __launch_bounds__(256)
void attn_combine_kernel(const float* __restrict__ mseg,
                         const float* __restrict__ sseg,
                         const float* __restrict__ accseg,
                         float* __restrict__ attn,
                         float* __restrict__ wc) {
  const int b   = blockIdx.x;
  const int tid = threadIdx.x;

  float M = mseg[b * SEG];
  #pragma unroll
  for (int s2 = 1; s2 < SEG; ++s2) M = fmaxf(M, mseg[b * SEG + s2]);
  float Z = 0.f;
  float w[SEG];
  #pragma unroll
  for (int s2 = 0; s2 < SEG; ++s2) {
    w[s2] = __expf(mseg[b * SEG + s2] - M);
    Z += sseg[b * SEG + s2] * w[s2];
  }
  const float invZ = 1.f / Z;

  float4 v0 = {0.f, 0.f, 0.f, 0.f};
  float4 v1 = {0.f, 0.f, 0.f, 0.f};
  #pragma unroll
  for (int s2 = 0; s2 < SEG; ++s2) {
    const float4* a4 = (const float4*)(accseg + (size_t)(b * SEG + s2) * D_);
    const float4 c0 = a4[tid];
    const float4 c1 = a4[256 + tid];
    v0.x += w[s2] * c0.x; v0.y += w[s2] * c0.y; v0.z += w[s2] * c0.z; v0.w += w[s2] * c0.w;
    v1.x += w[s2] * c1.x; v1.y += w[s2] * c1.y; v1.z += w[s2] * c1.z; v1.w += w[s2] * c1.w;
  }
  float4* wc4 = (float4*)(wc + (size_t)b * D_);
  v0.x *= invZ; v0.y *= invZ; v0.z *= invZ; v0.w *= invZ;
  v1.x *= invZ; v1.y *= invZ; v1.z *= invZ; v1.w *= invZ;
  wc4[tid]       = v0;
  wc4[256 + tid] = v1;

  for (int j = tid; j < L_; j += 256) {
    const size_t idx = (size_t)b * L_ + j;
    attn[idx] = __expf(attn[idx] - M) * invZ;  // masked: exp(-1e12-M) -> 0
  }
}

// ---------------------------------------------------------------------------
// K3: h_tilde[64,2048] = tanh( [wc | input][64,4096] @ W_out[2048,4096]^T )
// ---------------------------------------------------------------------------
__global__ __launch_bounds__(32)
void gemm_out_kernel(const float* __restrict__ wc,
                     const float* __restrict__ input,
                     const float* __restrict__ Wout,
                     float* __restrict__ hout) {
  const int lane = threadIdx.x;
  const int mn   = lane & 15;
  const int half = lane >> 4;
  const int n0   = blockIdx.x * 16;
  const int m0   = blockIdx.y * 16;

  const float* a_lo = wc    + (size_t)(m0 + mn) * D_;
  const float* a_hi = input + (size_t)(m0 + mn) * D_;
  const float* brow = Wout  + (size_t)(n0 + mn) * (2 * D_);

  v8f c = {};
  #pragma unroll 4
  for (int k0 = 0; k0 < 2 * D_; k0 += 4) {
    const int k = k0 + 2 * half;                 // pair (k,k+1) never straddles D_
    const float* ap = (k < D_) ? (a_lo + k) : (a_hi + (k - D_));
    v2f a = *(const v2f*)ap;
    v2f b = *(const v2f*)(brow + k);
    c = __builtin_amdgcn_wmma_f32_16x16x4_f32(false, a, false, b,
                                              (short)0, c, false, false);
  }
  #pragma unroll
  for (int r = 0; r < 8; ++r)
    hout[(size_t)(m0 + r + 8 * half) * D_ + n0 + mn] = tanhf(c[r]);
}

// ---------------------------------------------------------------------------
extern "C" void kernel_launch(void* const* d_in, const int* in_sizes, int n_in,
                              void* d_out, int out_size, void* d_ws, size_t ws_size,
                              hipStream_t stream) {
  const float*   input = (const float*)d_in[0];    // [B,D]
  const float*   ctx   = (const float*)d_in[1];    // [B,L,D]
  const uint8_t* mask  = (const uint8_t*)d_in[2];  // [B,L] bool
  const float*   Win   = (const float*)d_in[3];    // [D,D]
  const float*   Wout  = (const float*)d_in[4];    // [D,2D]

  float* hout = (float*)d_out;                     // [B,D]
  float* attn = (float*)d_out + (size_t)B_ * D_;   // [B,L]

  float* tgt    = (float*)d_ws;                    // B*D
  float* wc     = tgt    + (size_t)B_ * D_;        // B*D
  float* mseg   = wc     + (size_t)B_ * D_;        // B*SEG
  float* sseg   = mseg   + B_ * SEG;               // B*SEG
  float* accseg = sseg   + B_ * SEG;               // B*SEG*D

  gemm_in_kernel<<<dim3(D_ / 16, B_ / 16), 32, 0, stream>>>(input, Win, tgt);
  attn_partial_kernel<<<dim3(SEG, B_), 256, 0, stream>>>(ctx, mask, tgt, attn,
                                                         mseg, sseg, accseg);
  attn_combine_kernel<<<B_, 256, 0, stream>>>(mseg, sseg, accseg, attn, wc);
  gemm_out_kernel<<<dim3(D_ / 16, B_ / 16), 32, 0, stream>>>(wc, input, Wout, hout);
}